// SemanticInjectingPatchEmbed_74938589380664
// MI455X (gfx1250) — compile-verified
//
#include <hip/hip_runtime.h>
#include <hip/hip_bf16.h>

typedef float v2f __attribute__((ext_vector_type(2)));
typedef float v4f __attribute__((ext_vector_type(4)));
typedef float v8f __attribute__((ext_vector_type(8)));
typedef int   v4i __attribute__((ext_vector_type(4)));

#define BQ      64
#define NCQ     20
#define PQ      4
#define HWQ     224
#define EDQ     128
#define HPQ     56
#define NPATCH  (HPQ * HPQ)      // 3136 patches per image
#define KQ      64               // 4 channels * 16 pixels
#define MT      32               // patches per block
#define PITCH_A 68               // As row pitch (floats); 68*4 % 16 == 0 -> b128-aligned rows
#define PITCH_B 136              // Bs row pitch (floats) -> WMMA halves hit disjoint banks
#define PITCH_D 132              // Ds row pitch (floats); 132*4 % 16 == 0 -> b128-aligned rows

__launch_bounds__(256)
__global__ void sem_patch_embed_kernel(const float* __restrict__ rgb,
                                       const int*   __restrict__ mask,
                                       const float* __restrict__ proj_w,
                                       const float* __restrict__ proj_b,
                                       const float* __restrict__ ln_gamma,
                                       const float* __restrict__ ln_beta,
                                       const float* __restrict__ sem_emb,
                                       const float* __restrict__ cw,
                                       float*       __restrict__ out)
{
    __shared__ float Bs[KQ * PITCH_B];    // proj_w^T : [k][ed]
    __shared__ float As[MT * PITCH_A];    // gathered patch inputs : [p][k]
    __shared__ float Ds[MT * PITCH_D];    // GEMM result : [p][ed]
    __shared__ float s_mean[MT];
    __shared__ float s_rstd[MT];
    __shared__ int   s_cls[MT];

    const int t   = threadIdx.x;
    const int gp0 = blockIdx.x * MT;      // first global patch of this block

    // ---- stage B = proj_w^T into LDS: b128 global reads, transposed scalar LDS scatter ----
    {
        const v4f* pw4 = (const v4f*)proj_w;
        for (int idx4 = t; idx4 < (EDQ * KQ) / 4; idx4 += 256) {   // 2048 quads, 8 iters
            int ed = idx4 >> 4;                 // quad index / 16 (16 quads per ed row)
            int k0 = (idx4 & 15) << 2;          // first k of the quad
            v4f v  = pw4[idx4];
            Bs[(k0 + 0) * PITCH_B + ed] = v.x;
            Bs[(k0 + 1) * PITCH_B + ed] = v.y;
            Bs[(k0 + 2) * PITCH_B + ed] = v.z;
            Bs[(k0 + 3) * PITCH_B + ed] = v.w;
        }
    }

    // ---- gather A as b128 rows: one float4/int4 per (patch, channel, pixel-row) ----
    // K = c*16 + ph*4 + pw ; c==3 -> mask. Rows are 16B-aligned (x = wp*4).
    for (int row = t; row < MT * 16; row += 256) {                 // 512 rows, 2 iters
        int p   = row >> 4;
        int cph = row & 15;
        int c   = cph >> 2;
        int ph  = cph & 3;
        int gp  = gp0 + p;
        int b   = gp / NPATCH;
        int pi  = gp - b * NPATCH;
        int hp  = pi / HPQ;
        int wp  = pi - hp * HPQ;
        int y   = hp * PQ + ph;
        int x   = wp * PQ;
        v4f v;
        if (c < 3) {
            v = *(const v4f*)&rgb[(((size_t)b * 3 + c) * HWQ + y) * HWQ + x];
        } else {
            v4i mi = *(const v4i*)&mask[((size_t)b * HWQ + y) * HWQ + x];
            v.x = (float)mi.x; v.y = (float)mi.y; v.z = (float)mi.z; v.w = (float)mi.w;
        }
        *(v4f*)&As[p * PITCH_A + c * 16 + ph * 4] = v;
    }
    __syncthreads();

    // ---- WMMA FP32 GEMM: wave w owns ED columns [w*16, w*16+16), 2 M-subtiles ----
    const int wave = t >> 5;              // 0..7
    const int lane = t & 31;
    const int half = lane >> 4;           // 0: K pair {0,1}, 1: K pair {2,3}
    const int l15  = lane & 15;
    const int ed   = wave * 16 + l15;     // output column for B/C/D fragments

    v8f acc0 = {0.f, 0.f, 0.f, 0.f, 0.f, 0.f, 0.f, 0.f};
    v8f acc1 = {0.f, 0.f, 0.f, 0.f, 0.f, 0.f, 0.f, 0.f};

    #pragma unroll
    for (int kk = 0; kk < KQ; kk += 4) {
        const int k0 = kk + half * 2;
        v2f bfrag, a0, a1;
        bfrag.x = Bs[(k0 + 0) * PITCH_B + ed];
        bfrag.y = Bs[(k0 + 1) * PITCH_B + ed];
        a0.x = As[l15 * PITCH_A + k0 + 0];
        a0.y = As[l15 * PITCH_A + k0 + 1];
        a1.x = As[(16 + l15) * PITCH_A + k0 + 0];
        a1.y = As[(16 + l15) * PITCH_A + k0 + 1];
        acc0 = __builtin_amdgcn_wmma_f32_16x16x4_f32(false, a0, false, bfrag,
                                                     (short)0, acc0, false, false);
        acc1 = __builtin_amdgcn_wmma_f32_16x16x4_f32(false, a1, false, bfrag,
                                                     (short)0, acc1, false, false);
    }

    // ---- bias, spill D tiles to LDS ----
    const float bias = proj_b[ed];
    #pragma unroll
    for (int r = 0; r < 8; ++r) {
        Ds[(r + half * 8) * PITCH_D + ed]      = acc0[r] + bias;  // patches 0..15
        Ds[(16 + r + half * 8) * PITCH_D + ed] = acc1[r] + bias;  // patches 16..31
    }
    __syncthreads();

    // ---- per-patch weighted-histogram argmax (threads 0..31, register-only) ----
    if (t < MT) {
        int lbl[16];
        #pragma unroll
        for (int j = 0; j < 16; ++j)
            lbl[j] = (int)As[t * PITCH_A + 48 + j];   // c==3 slice = mask pixels
        float best = -1.0f;
        int   bc   = 0;
        #pragma unroll
        for (int c = 0; c < NCQ; ++c) {
            int cnt = 0;
            #pragma unroll
            for (int j = 0; j < 16; ++j) cnt += (lbl[j] == c) ? 1 : 0;
            float s = (float)cnt * cw[c];
            if (s > best) { best = s; bc = c; }       // strict > keeps first index (argmax tie rule)
        }
        s_cls[t] = bc;
    }

    // ---- LayerNorm stats: 8 lanes per patch, b128 LDS reads, shuffle-tree reduce ----
    {
        const int p     = t >> 3;         // 0..31
        const int chunk = t & 7;
        float s = 0.f, s2 = 0.f;
        #pragma unroll
        for (int j = 0; j < 4; ++j) {
            v4f v = *(const v4f*)&Ds[p * PITCH_D + chunk * 16 + j * 4];
            s  += v.x + v.y + v.z + v.w;
            s2 += v.x * v.x + v.y * v.y + v.z * v.z + v.w * v.w;
        }
        #pragma unroll
        for (int off = 4; off >= 1; off >>= 1) {
            s  += __shfl_xor(s,  off, 8);
            s2 += __shfl_xor(s2, off, 8);
        }
        if (chunk == 0) {
            float mean = s * (1.0f / EDQ);
            float var  = s2 * (1.0f / EDQ) - mean * mean;
            s_mean[p] = mean;
            s_rstd[p] = rsqrtf(var + 1e-5f);
        }
    }
    __syncthreads();

    // ---- fused LN + semantic add: float4 everywhere, non-temporal b128 streaming stores ----
    {
        const v4f* g4 = (const v4f*)ln_gamma;
        const v4f* b4 = (const v4f*)ln_beta;
        const v4f* s4 = (const v4f*)sem_emb;
        for (int idx4 = t; idx4 < (MT * EDQ) / 4; idx4 += 256) {   // 1024 quads, 4 iters
            int p  = idx4 >> 5;
            int e4 = idx4 & 31;
            v4f v  = *(const v4f*)&Ds[p * PITCH_D + e4 * 4];
            v4f gm = g4[e4];
            v4f bt = b4[e4];
            v4f sm = s4[s_cls[p] * (EDQ / 4) + e4];
            float mean = s_mean[p];
            float rstd = s_rstd[p];
            v4f o;
            o.x = (v.x - mean) * rstd * gm.x + bt.x + sm.x;
            o.y = (v.y - mean) * rstd * gm.y + bt.y + sm.y;
            o.z = (v.z - mean) * rstd * gm.z + bt.z + sm.z;
            o.w = (v.w - mean) * rstd * gm.w + bt.w + sm.w;
            size_t gp = (size_t)(gp0 + p);
            __builtin_nontemporal_store(o, (v4f*)&out[gp * EDQ + e4 * 4]);
        }
    }
}

extern "C" void kernel_launch(void* const* d_in, const int* in_sizes, int n_in,
                              void* d_out, int out_size, void* d_ws, size_t ws_size,
                              hipStream_t stream) {
    const float* rgb      = (const float*)d_in[0];
    const int*   mask     = (const int*)  d_in[1];
    const float* proj_w   = (const float*)d_in[2];
    const float* proj_b   = (const float*)d_in[3];
    const float* ln_gamma = (const float*)d_in[4];
    const float* ln_beta  = (const float*)d_in[5];
    const float* sem_emb  = (const float*)d_in[6];
    const float* cw       = (const float*)d_in[7];
    float* out = (float*)d_out;

    const int total_patches = BQ * NPATCH;          // 200704
    const int blocks = total_patches / MT;          // 6272
    sem_patch_embed_kernel<<<blocks, 256, 0, stream>>>(
        rgb, mask, proj_w, proj_b, ln_gamma, ln_beta, sem_emb, cw, out);
}